// Softmax_59485297049782
// MI455X (gfx1250) — compile-verified
//
#include <hip/hip_runtime.h>
#include <stdint.h>

typedef float v4f __attribute__((ext_vector_type(4)));

#define THREADS 256          // 8 wave32s
#define COLS    2048         // row length (fixed by reference)
#define RPB     8            // rows per block
#define NWAVES  (THREADS / 32)

// Each wave issues 2 async b128 loads per row (lane t covers bytes [t*16) and
// [t*16 + 4096)).  INST_OFFSET applies to both the global and the LDS address,
// so the same VGPR pair describes both chunks.  The input is a 256 MB
// single-use stream -> non-temporal hint keeps it from displacing L2.
__device__ __forceinline__ void async_row_load(uint64_t gbase, uint32_t ldsoff, uint32_t goff) {
    asm volatile(
        "global_load_async_to_lds_b128 %0, %1, %2 th:TH_LOAD_NT\n\t"
        "global_load_async_to_lds_b128 %0, %1, %2 offset:4096 th:TH_LOAD_NT"
        :
        : "v"(ldsoff), "v"(goff), "s"(gbase)
        : "memory");
}

__global__ __launch_bounds__(THREADS)
void ckks_softmax_async(const float* __restrict__ x,
                        const float* __restrict__ ec_g,   // 9 exp coeffs (power basis)
                        const float* __restrict__ ic_g,   // 5 inverse coeffs
                        const int*   __restrict__ itp,    // newton iterations
                        float*       __restrict__ out,
                        int rows)
{
    __shared__ __align__(16) float buf[2][COLS];   // double buffer: 2 x 8 KB
    __shared__ float wsum[NWAVES];

    const int tid = threadIdx.x;

    // Broadcast-load small coefficient arrays into registers (uniform -> SMEM loads).
    float ec[9], ic[5];
#pragma unroll
    for (int i = 0; i < 9; ++i) ec[i] = ec_g[i];
#pragma unroll
    for (int i = 0; i < 5; ++i) ic[i] = ic_g[i];
    const int iters = itp[0];

    const int row0 = blockIdx.x * RPB;
    if (row0 >= rows) return;
    const int rend = (row0 + RPB < rows) ? (row0 + RPB) : rows;

    // Block-relative LDS byte offsets of this lane's 16-B chunk in each buffer.
    const uint32_t goff = (uint32_t)tid * 16u;
    const uint32_t l0 = (uint32_t)(uintptr_t)(&buf[0][0]) + goff;
    const uint32_t l1 = (uint32_t)(uintptr_t)(&buf[1][0]) + goff;

    // Prefetch first row into buffer 0.
    async_row_load((uint64_t)(uintptr_t)(x + (size_t)row0 * COLS), l0, goff);

    for (int r = row0; r < rend; ++r) {
        const int par = (r - row0) & 1;

        if (r + 1 < rend) {
            // Prefetch next row into the other buffer, then wait until only
            // those 2 loads remain outstanding (async loads complete in order).
            async_row_load((uint64_t)(uintptr_t)(x + (size_t)(r + 1) * COLS),
                           par ? l0 : l1, goff);
            asm volatile("s_wait_asynccnt 0x2" ::: "memory");
        } else {
            asm volatile("s_wait_asynccnt 0x0" ::: "memory");
        }

        // Read this lane's 8 elements of row r from LDS (two ds_load_b128).
        const v4f* bp = (const v4f*)(&buf[par][0]);
        v4f xa = bp[tid];
        v4f xb = bp[tid + THREADS];

        // e = degree-8 Horner, kept in registers; accumulate partial row sum.
        float e[8];
        float psum = 0.0f;
#pragma unroll
        for (int k = 0; k < 8; ++k) {
            float xv = (k < 4) ? xa[k] : xb[k - 4];
            float acc = ec[8];
#pragma unroll
            for (int j = 7; j >= 0; --j) acc = __builtin_fmaf(acc, xv, ec[j]);
            e[k] = acc;
            psum += acc;
        }

        // wave32 butterfly reduction, then cross-wave via LDS.
#pragma unroll
        for (int off = 16; off > 0; off >>= 1)
            psum += __shfl_xor(psum, off, 32);
        if ((tid & 31) == 0) wsum[tid >> 5] = psum;
        __syncthreads();
        float s = 0.0f;
#pragma unroll
        for (int w = 0; w < NWAVES; ++w) s += wsum[w];

        // Initial reciprocal guess (degree-4 Horner) + Newton refinement.
        float y = ic[4];
#pragma unroll
        for (int j = 3; j >= 0; --j) y = __builtin_fmaf(y, s, ic[j]);
        for (int i = 0; i < iters; ++i) y = y * (2.0f - s * y);

        // out = e * y, streamed non-temporally (no reuse of the 256 MB output).
        v4f o0, o1;
#pragma unroll
        for (int k = 0; k < 4; ++k) { o0[k] = e[k] * y; o1[k] = e[k + 4] * y; }
        v4f* op = (v4f*)(out + (size_t)r * COLS);
        __builtin_nontemporal_store(o0, op + tid);
        __builtin_nontemporal_store(o1, op + tid + THREADS);

        __syncthreads();   // wsum is rewritten next iteration
    }
}

extern "C" void kernel_launch(void* const* d_in, const int* in_sizes, int n_in,
                              void* d_out, int out_size, void* d_ws, size_t ws_size,
                              hipStream_t stream) {
    const float* x  = (const float*)d_in[0];
    const float* ec = (const float*)d_in[1];
    const float* ic = (const float*)d_in[2];
    const int*   it = (const int*)d_in[3];
    float* out = (float*)d_out;

    const int rows = in_sizes[0] / COLS;           // 16 * 2048 = 32768
    const int grid = (rows + RPB - 1) / RPB;       // 4096 blocks

    ckks_softmax_async<<<grid, THREADS, 0, stream>>>(x, ec, ic, it, out, rows);
}